// SplatterModel_79302276153346
// MI455X (gfx1250) — compile-verified
//
#include <hip/hip_runtime.h>
#include <math.h>

typedef float v2f __attribute__((ext_vector_type(2)));
typedef float v8f __attribute__((ext_vector_type(8)));

#define GN      1000          // number of gaussians
#define GNP     1008          // padded to 63 * 16
#define NTILES  63
#define SEPS    1e-6f
#define LOG2PI  1.8378770664093453f

__device__ __forceinline__ float sigmoidf_(float x) { return 1.0f / (1.0f + __expf(-x)); }

// ---------------------------------------------------------------------------
// Phase 0: per-gaussian coefficient precompute.
// kernel_log[n](x,y) = q0*x^2 + q1*x*y + q2*y^2 + q3*x + q4*y + q5
// Stored as K-pairs (SoA of float2) matching the WMMA 16x4 f32 A-matrix
// layout: lanes 0-15 read pair (q0,q1) [K=0,1], lanes 16-31 pair (q2,q3)
// [K=2,3]; second WMMA uses pairs (q4,q5) / (0,0).
// ---------------------------------------------------------------------------
__global__ void splat_precompute(const float* __restrict__ rho,
                                 const float* __restrict__ sigma,
                                 const float* __restrict__ coords,
                                 const float* __restrict__ alpha,
                                 const float* __restrict__ colors,
                                 float2* __restrict__ Qp,   // [4][GNP] K-pairs
                                 float4* __restrict__ AC)   // [GNP] {alpha,r,g,b}
{
    int n = blockIdx.x * blockDim.x + threadIdx.x;
    if (n >= GNP) return;

    float q0 = 0.f, q1 = 0.f, q2 = 0.f, q3 = 0.f, q4 = 0.f, q5 = -1e30f;
    float aa = 0.f, cr = 0.f, cg = 0.f, cb = 0.f;

    if (n < GN) {
        float r  = sigmoidf_(rho[n]) + SEPS;
        float sx = tanhf(sigma[2 * n + 0] * 0.5f) + 1e-4f;
        float sy = tanhf(sigma[2 * n + 1] * 0.5f) + 1e-4f;

        float a = sx * sx + SEPS;
        float b = sx * sy * r;
        float c = sy * sy + SEPS;
        float det = a * c - b * b;
        float inv_det = 1.0f / det;

        // M = -0.5 * inv(cov)
        float mxx = -0.5f * c * inv_det;
        float mxy =  0.5f * b * inv_det;
        float myy = -0.5f * a * inv_det;

        float cx = tanhf(coords[2 * n + 0]) - 0.5f;
        float cy = tanhf(coords[2 * n + 1]) - 0.5f;

        q0 = mxx;
        q1 = 2.0f * mxy;
        q2 = myy;
        q3 = 2.0f * (mxx * cx + mxy * cy);
        q4 = 2.0f * (myy * cy + mxy * cx);
        q5 = mxx * cx * cx + 2.0f * mxy * cx * cy + myy * cy * cy
             - LOG2PI + 0.5f * logf(det + SEPS);

        aa = sigmoidf_(alpha[n]);
        cr = sigmoidf_(colors[3 * n + 0]);
        cg = sigmoidf_(colors[3 * n + 1]);
        cb = sigmoidf_(colors[3 * n + 2]);
    }

    Qp[0 * GNP + n] = make_float2(q0, q1);
    Qp[1 * GNP + n] = make_float2(q2, q3);
    Qp[2 * GNP + n] = make_float2(q4, q5);
    Qp[3 * GNP + n] = make_float2(0.f, 0.f);
    AC[n] = make_float4(aa, cr, cg, cb);
}

// ---------------------------------------------------------------------------
// Phase 1: streaming WMMA splat with online softmax.
// One wave32 owns 16 pixels; loops over 63 tiles of 16 gaussians.
// A (16x4, M=gaussian) from LDS, B (4x16, N=pixel) built in registers.
// D[v] per lane = gaussian (v + 8*half), pixel (lane & 15); pixel state
// replicated across lane pairs (l, l+16), merged via __shfl_xor(.,16).
// ---------------------------------------------------------------------------
__global__ void __launch_bounds__(256)
splat_render(const float* __restrict__ xy,
             const float2* __restrict__ Qp,
             const float4* __restrict__ AC,
             float* __restrict__ out)
{
    __shared__ float2 Qs[4 * GNP];   // 32256 B
    __shared__ float4 ACs[GNP];      // 16128 B

    int tid = threadIdx.x;
    for (int i = tid; i < 4 * GNP; i += 256) Qs[i] = Qp[i];
    for (int i = tid; i < GNP;     i += 256) ACs[i] = AC[i];
    __syncthreads();

    int lane = tid & 31;
    int wave = tid >> 5;
    int h    = lane >> 4;                    // half-wave select
    int ln   = lane & 15;
    int mp   = (blockIdx.x * 8 + wave) * 16; // 16 pixels per wave
    int p    = mp + ln;

    float x = xy[2 * p + 0];
    float y = xy[2 * p + 1];

    // B-matrix (4x16 f32, N=lane&15): VGPR0 = K(0 or 2), VGPR1 = K(1 or 3)
    v2f B0, B1;
    B0.x = h ? y * y : x * x;     // K=2 : K=0
    B0.y = h ? x     : x * y;     // K=3 : K=1
    B1.x = h ? 0.f   : y;         // K=6 : K=4
    B1.y = h ? 0.f   : 1.0f;      // K=7 : K=5

    float run_m = -1e30f;
    float sa = 0.f, sr = 0.f, sg = 0.f, sb = 0.f;

    for (int t = 0; t < NTILES; ++t) {
        int g = t * 16 + ln;
        float2 a0p = Qs[h * GNP + g];        // (q0,q1) or (q2,q3)
        float2 a1p = Qs[(2 + h) * GNP + g];  // (q4,q5) or (0,0)
        v2f A0; A0.x = a0p.x; A0.y = a0p.y;
        v2f A1; A1.x = a1p.x; A1.y = a1p.y;

        v8f z = {};
        z = __builtin_amdgcn_wmma_f32_16x16x4_f32(false, A1, false, B1,
                                                  (short)0, z, false, false);
        z = __builtin_amdgcn_wmma_f32_16x16x4_f32(false, A0, false, B0,
                                                  (short)0, z, false, false);

        // tile max over this lane's 8 gaussians, merged across lane halves
        float tmax = z[0];
        #pragma unroll
        for (int i = 1; i < 8; ++i) tmax = fmaxf(tmax, z[i]);
        tmax = fmaxf(tmax, __shfl_xor(tmax, 16, 32));

        float nm = fmaxf(run_m, tmax);
        float sc = __expf(run_m - nm);       // rescale old accumulators

        float pa = 0.f, pr = 0.f, pg = 0.f, pb = 0.f;
        int gb = t * 16 + h * 8;
        #pragma unroll
        for (int i = 0; i < 8; ++i) {
            float  e  = __expf(z[i] - nm);
            float4 ac = ACs[gb + i];
            pa = fmaf(e, ac.x, pa);
            pr = fmaf(e, ac.y, pr);
            pg = fmaf(e, ac.z, pg);
            pb = fmaf(e, ac.w, pb);
        }
        pa += __shfl_xor(pa, 16, 32);
        pr += __shfl_xor(pr, 16, 32);
        pg += __shfl_xor(pg, 16, 32);
        pb += __shfl_xor(pb, 16, 32);

        sa = fmaf(sa, sc, pa);
        sr = fmaf(sr, sc, pr);
        sg = fmaf(sg, sc, pg);
        sb = fmaf(sb, sc, pb);
        run_m = nm;
    }

    if (h == 0) {
        float inv = 1.0f / (sa + SEPS);
        out[3 * p + 0] = sr * inv;
        out[3 * p + 1] = sg * inv;
        out[3 * p + 2] = sb * inv;
    }
}

// ---------------------------------------------------------------------------
extern "C" void kernel_launch(void* const* d_in, const int* in_sizes, int n_in,
                              void* d_out, int out_size, void* d_ws, size_t ws_size,
                              hipStream_t stream)
{
    const float* rho    = (const float*)d_in[0];
    const float* sigma  = (const float*)d_in[1];
    const float* coords = (const float*)d_in[2];
    const float* alpha  = (const float*)d_in[3];
    const float* colors = (const float*)d_in[4];
    const float* xy     = (const float*)d_in[5];
    float*       out    = (float*)d_out;

    float2* Qp = (float2*)d_ws;                                    // 32256 B
    float4* AC = (float4*)((char*)d_ws + 4 * GNP * sizeof(float2)); // +16128 B

    splat_precompute<<<(GNP + 255) / 256, 256, 0, stream>>>(
        rho, sigma, coords, alpha, colors, Qp, AC);

    // 65536 pixels / (8 waves * 16 px) = 512 blocks
    splat_render<<<512, 256, 0, stream>>>(xy, Qp, AC, out);
}